// StreamingKVCache_81844896792692
// MI455X (gfx1250) — compile-verified
//
#include <hip/hip_runtime.h>
#include <hip/hip_bf16.h>
#include <stdint.h>

// Streaming KV-cache rebuild + RoPE for MI455X (gfx1250, wave32).
// Pure bandwidth problem: ~0.55 GB traffic -> ~24us roofline at 23.3 TB/s.
// No GEMM -> WMMA inapplicable; optimize the memory path instead:
//  - 2D grid (token, batch): no integer-division SALU blob for block indices
//  - one 128-thread block per (b, token): 64 lanes build the RoPE sin/cos
//    table once in LDS, all 8 heads reuse it (8x fewer transcendentals)
//  - K plane: b128 loads, register RoPE, non-temporal b128 stores
//  - V plane: gfx1250 async DMA  global -> LDS -> global
//    (global_load_async_to_lds_b128 / global_store_async_from_lds_b128,
//     synchronized with one s_wait_asynccnt 0); data never touches VGPRs
//  - global_prefetch on the K source row issued before the sincos+barrier
//    section, overlapping transcendental VALU with memory latency

#define PAGE_SZ   16
#define SBUDGET   4096
#define NUM_SINK  4
#define HEADS     8
#define HDIM      128
// ln(10000)
#define LN_THETA  9.2103403719761836f

// Native clang vector so nontemporal builtins accept it; lowers to b128 ops.
typedef float __attribute__((ext_vector_type(4))) f4;

__global__ __launch_bounds__(128)
void StreamingKVCache_kernel(const float* __restrict__ kv_cache,
                             const float* __restrict__ knew,
                             const float* __restrict__ vnew,
                             const int*   __restrict__ query_lens,
                             const int*   __restrict__ ctx_ptr,
                             float*       __restrict__ out,
                             int max_kv, int seq_len)
{
    __shared__ __align__(16) float s_cos[64];
    __shared__ __align__(16) float s_sin[64];
    // V-plane staging for the async global->LDS->global DMA copy:
    // slot [tid] = first half (d0), slot [128+tid] = second half (d0+64).
    __shared__ __align__(16) f4 s_vstage[256];

    const int tid = threadIdx.x;
    const int p   = blockIdx.x;       // destination token index within batch
    const int b   = blockIdx.y;       // batch

    // Uniform scalars (SALU path)
    const int ctx = *ctx_ptr;
    const int ql  = query_lens[b];

    const bool active  = ql > 0;
    const bool rolling = active && (ctx + ql >  SBUDGET);
    const bool plain   = active && (ctx + ql <= SBUDGET);

    // ---- source selection (block-uniform) ----
    int  src_tok = p;     // default: pass-through from kv_cache
    bool src_new = false; // true: read from k/v tensors
    if (rolling && p >= NUM_SINK && p < SBUDGET - ql) {
        const int upper = (ctx < SBUDGET) ? ctx : SBUDGET;
        const int srs   = upper - (SBUDGET - ql - NUM_SINK);
        int t = p + srs - NUM_SINK;
        t = t < 0 ? 0 : (t > max_kv - 1 ? max_kv - 1 : t);
        src_tok = t; src_new = false;
    } else if (rolling && p >= SBUDGET - ql && p < SBUDGET) {
        int t = p - (SBUDGET - ql);
        t = t < 0 ? 0 : (t > seq_len - 1 ? seq_len - 1 : t);
        src_tok = t; src_new = true;
    } else if (plain && p >= ctx && p < ctx + ql) {
        int t = p - ctx;
        t = t < 0 ? 0 : (t > seq_len - 1 ? seq_len - 1 : t);
        src_tok = t; src_new = true;
    }

    // ---- per-thread lane mapping: head h, float4 column c in first half ----
    const int h  = tid >> 4;   // 0..7
    const int c  = tid & 15;   // 0..15
    const int d0 = c * 4;      // element offset in [0,64): paired with d0+64

    const int PLANE = PAGE_SZ * HEADS * HDIM;   // one (page, kv) plane in floats

    const float *kA, *kB, *vA, *vB;
    if (src_new) {
        const size_t base = ((size_t)(b * seq_len + src_tok) * HEADS + h) * HDIM;
        kA = knew + base + d0;  kB = kA + 64;
        vA = vnew + base + d0;  vB = vA + 64;
    } else {
        const int    g    = b * max_kv + src_tok;
        const size_t page = (size_t)(g >> 4) * (size_t)(2 * PLANE);
        const size_t off  = (size_t)(g & 15) * (HEADS * HDIM) + (size_t)h * HDIM + d0;
        kA = kv_cache + page + off;          kB = kA + 64;   // kv=0 plane
        vA = kv_cache + page + PLANE + off;  vB = vA + 64;   // kv=1 plane
    }

    // Generic LDS pointer = {aperture_hi32, lds_offset32}: low 32 bits are
    // the DS address the async instructions take in their LDS-address VGPR.
    const uint32_t ldsA = (uint32_t)(uintptr_t)&s_vstage[tid];
    const uint32_t ldsB = (uint32_t)(uintptr_t)&s_vstage[128 + tid];
    const uint64_t gvA  = (uint64_t)(uintptr_t)vA;
    const uint64_t gvB  = (uint64_t)(uintptr_t)vB;

    // Kick off the V-plane copy on the async DMA path (ASYNCcnt += 2).
    // Data flows global -> LDS without touching VGPRs.
    asm volatile("global_load_async_to_lds_b128 %0, %1, off"
                 :: "v"(ldsA), "v"(gvA) : "memory");
    asm volatile("global_load_async_to_lds_b128 %0, %1, off"
                 :: "v"(ldsB), "v"(gvB) : "memory");

    // gfx1250: pull the K source line toward the caches while the sincos
    // table is computed below. (global_prefetch_b8)
    __builtin_prefetch(kA, 0, 3);

    // ---- RoPE angle table for this token (only first SBUDGET positions) ----
    const bool do_rope = (p < SBUDGET);   // block-uniform
    if (do_rope && tid < 64) {
        // inv_freq[j] = theta^(-j/64);  ang = p * inv_freq[j]
        const float invf = __builtin_expf(-(float)tid * (LN_THETA / 64.0f));
        const float ang  = (float)p * invf;
        s_cos[tid] = __builtin_cosf(ang);
        s_sin[tid] = __builtin_sinf(ang);
    }
    __syncthreads();

    // ---- K plane: load, rotate, streaming store ----
    const f4 ka = *(const f4*)kA;
    const f4 kb = *(const f4*)kB;

    f4 oA, oB;
    if (do_rope) {
        const f4 co = *(const f4*)&s_cos[d0];   // ds_load_b128
        const f4 si = *(const f4*)&s_sin[d0];
        oA.x = fmaf(ka.x, co.x, -kb.x * si.x);
        oA.y = fmaf(ka.y, co.y, -kb.y * si.y);
        oA.z = fmaf(ka.z, co.z, -kb.z * si.z);
        oA.w = fmaf(ka.w, co.w, -kb.w * si.w);
        oB.x = fmaf(ka.x, si.x,  kb.x * co.x);
        oB.y = fmaf(ka.y, si.y,  kb.y * co.y);
        oB.z = fmaf(ka.z, si.z,  kb.z * co.z);
        oB.w = fmaf(ka.w, si.w,  kb.w * co.w);
    } else {
        oA = ka; oB = kb;
    }

    const int    go    = b * max_kv + p;
    const size_t opage = (size_t)(go >> 4) * (size_t)(2 * PLANE);
    const size_t ooff  = (size_t)(go & 15) * (HEADS * HDIM) + (size_t)h * HDIM + d0;
    float* okA = out + opage + ooff;            // kv=0 plane
    float* ovA = out + opage + PLANE + ooff;    // kv=1 plane

    // Stream K output past the caches (TH=NT stores).
    __builtin_nontemporal_store(oA, (f4*)okA);
    __builtin_nontemporal_store(oB, (f4*)(okA + 64));

    // ---- V plane: drain async loads, then async-store LDS -> global ----
    // Wave-local wait: each lane only touches its own staging slot, so
    // asynccnt==0 fully orders the LDS write before the LDS read below.
    const uint64_t goA = (uint64_t)(uintptr_t)ovA;
    const uint64_t goB = (uint64_t)(uintptr_t)(ovA + 64);
    asm volatile("s_wait_asynccnt 0x0" ::: "memory");
    asm volatile("global_store_async_from_lds_b128 %0, %1, off"
                 :: "v"(goA), "v"(ldsA) : "memory");
    asm volatile("global_store_async_from_lds_b128 %0, %1, off"
                 :: "v"(goB), "v"(ldsB) : "memory");
    // s_endpgm performs an implicit wait-idle, draining the async stores.
}

extern "C" void kernel_launch(void* const* d_in, const int* in_sizes, int n_in,
                              void* d_out, int out_size, void* d_ws, size_t ws_size,
                              hipStream_t stream) {
    (void)n_in; (void)out_size; (void)d_ws; (void)ws_size;

    const float* kv_cache   = (const float*)d_in[0];
    const float* k          = (const float*)d_in[1];
    const float* v          = (const float*)d_in[2];
    const int*   query_lens = (const int*)  d_in[3];
    const int*   ctx_ptr    = (const int*)  d_in[5];   // context_len scalar (device)
    float*       out        = (float*)d_out;

    // Shape constants fixed by the reference: H=8, D=128, PAGE=16.
    const int bsz          = in_sizes[3];                         // len(query_lens)
    const long total_toks  = (long)in_sizes[0] / (2 * HEADS * HDIM);
    const int  max_kv      = (int)(total_toks / bsz);             // 8192
    const int  seq_len     = in_sizes[1] / (bsz * HEADS * HDIM);  // 2048

    dim3 grid(max_kv, bsz);            // one block per (token, batch)
    StreamingKVCache_kernel<<<grid, 128, 0, stream>>>(
        kv_cache, k, v, query_lens, ctx_ptr, out, max_kv, seq_len);
}